// RandBox_67559835566444
// MI455X (gfx1250) — compile-verified
//
#include <hip/hip_runtime.h>

#define NMS_THR    0.7f
#define MIN_RATE   0.1f
#define MAX_FINAL  50
#define SORT_M     4096      // next pow2 >= N (N = 4000)
#define BLOCK_T    512       // 16 wave32s

#define AS1 __attribute__((address_space(1)))
#define AS3 __attribute__((address_space(3)))

typedef int v4i __attribute__((vector_size(16)));

// ---------------- CDNA5 async global->LDS staging (probe-guarded) -----------
#if defined(__HIP_DEVICE_COMPILE__)
#if __has_builtin(__builtin_amdgcn_global_load_async_to_lds_b128)
#define HAVE_ASYNC_LDS 1
__device__ __forceinline__ void async_copy16(const float4* g, float4* l) {
    // (global v4i*, lds v4i*, imm offset, imm cpol) per hipcc diagnostic
    __builtin_amdgcn_global_load_async_to_lds_b128(
        (AS1 v4i*)g, (AS3 v4i*)l, 0, 0);
}
#elif __has_builtin(__builtin_amdgcn_load_to_lds)
#define HAVE_ASYNC_LDS 1
__device__ __forceinline__ void async_copy16(const float4* g, float4* l) {
    __builtin_amdgcn_load_to_lds((AS1 v4i*)g, (AS3 v4i*)l, 16, 0, 0);
}
#endif
#endif

__device__ __forceinline__ void async_wait_all() {
#if __has_builtin(__builtin_amdgcn_s_wait_asynccnt)
    __builtin_amdgcn_s_wait_asynccnt(0);
#else
    asm volatile("s_wait_asynccnt 0" ::: "memory");
#endif
}

// IoU, torchvision convention (no +1), matches reference incl. epsilon
__device__ __forceinline__ float iou_f(float4 a, float4 b) {
    float areaA = (a.z - a.x) * (a.w - a.y);
    float areaB = (b.z - b.x) * (b.w - b.y);
    float xx1 = fmaxf(a.x, b.x), yy1 = fmaxf(a.y, b.y);
    float xx2 = fminf(a.z, b.z), yy2 = fminf(a.w, b.w);
    float inter = fmaxf(xx2 - xx1, 0.0f) * fmaxf(yy2 - yy1, 0.0f);
    return inter / (areaA + areaB - inter + 1e-9f);
}

// One workgroup per image. LDS: float4 sBox[N] (scaled boxes) + u64 sKey[4096]
__global__ __launch_bounds__(BLOCK_T)
void nms_per_image_kernel(const float* __restrict__ boxes_in,
                          const float* __restrict__ scores,
                          const int*   __restrict__ num_final,
                          const int*   __restrict__ img_shapes,
                          float*       __restrict__ out,   // [B*50*4 | B*50 | B]
                          int N, int B) {
    extern __shared__ unsigned char smem[];
    float4*             sBox = (float4*)smem;
    unsigned long long* sKey = (unsigned long long*)(smem + (size_t)N * sizeof(float4));

    const int img = blockIdx.x;
    const int tid = threadIdx.x;
    const float4* gBox = (const float4*)boxes_in + (size_t)img * N;

    // ---- Phase 1: stage raw boxes into LDS (async DMA path on gfx1250) ----
#ifdef HAVE_ASYNC_LDS
    for (int i = tid; i < N; i += BLOCK_T) async_copy16(&gBox[i], &sBox[i]);
    async_wait_all();
#else
    for (int i = tid; i < N; i += BLOCK_T) sBox[i] = gBox[i];
#endif
    __syncthreads();

    const float H = (float)img_shapes[img * 2 + 0];
    const float W = (float)img_shapes[img * 2 + 1];

    // ---- Phase 2: order corners, scale, validity; build packed sort keys ----
    // key = sortable_u32(valid ? -score : +inf) in high 32 bits, index in low.
    for (int i = tid; i < SORT_M; i += BLOCK_T) {
        unsigned su = 0xFF800000u;   // sortable(+inf): invalid / padding
        if (i < N) {
            float4 rb = sBox[i];
            float x1 = fminf(rb.x, rb.z) * W;
            float x2 = fmaxf(rb.x, rb.z) * W;
            float y1 = fminf(rb.y, rb.w) * H;
            float y2 = fmaxf(rb.y, rb.w) * H;
            sBox[i] = make_float4(x1, y1, x2, y2);
            bool valid = ((y2 - y1) > MIN_RATE * H) && ((x2 - x1) > MIN_RATE * W);
            if (valid) {
                float k = -scores[(size_t)img * N + i];
                unsigned fu = __float_as_uint(k);
                su = (fu & 0x80000000u) ? ~fu : (fu | 0x80000000u);
            }
        }
        sKey[i] = ((unsigned long long)su << 32) | (unsigned)i;
    }

    // ---- Phase 3: bitonic sort of 4096 packed keys in LDS (ascending) ----
    for (unsigned k = 2; k <= SORT_M; k <<= 1) {
        for (unsigned j = k >> 1; j > 0; j >>= 1) {
            __syncthreads();
            for (unsigned i = tid; i < SORT_M; i += BLOCK_T) {
                unsigned p = i ^ j;
                if (p > i) {
                    unsigned long long a = sKey[i];
                    unsigned long long b = sKey[p];
                    bool up = ((i & k) == 0);
                    if ((a > b) == up) { sKey[i] = b; sKey[p] = a; }
                }
            }
        }
    }
    __syncthreads();

    // ---- Phase 4: serial greedy NMS on wave 0 (wave32) ------------------
    // Each lane caches kept box for slot `lane` (kb0) and `lane+32` (kb1);
    // candidate vs all kept = one ballot. Early exit at num_final kept.
    if (tid < 32) {
        const int lane  = tid;
        const int limit = min(num_final[img], MAX_FINAL);
        float* outBoxes = out;
        float* outMask  = out + (size_t)B * MAX_FINAL * 4;
        float* outCnt   = outMask + (size_t)B * MAX_FINAL;

        float4 kb0 = make_float4(0.f, 0.f, 0.f, 0.f);
        float4 kb1 = make_float4(0.f, 0.f, 0.f, 0.f);
        int nk = 0;

        for (int c = 0; c < SORT_M && nk < limit; ++c) {
            unsigned long long sk = sKey[c];              // uniform LDS read
            unsigned ku = (unsigned)(sk >> 32);
            if (ku & 0x80000000u) break;                  // invalid/pad: done
            int bi = (int)(sk & 0xFFFFFFFFu);
            float4 cb = sBox[bi];                         // uniform gather

            int sup = 0;
            if (lane < nk       && iou_f(cb, kb0) > NMS_THR) sup = 1;
            if (lane + 32 < nk  && iou_f(cb, kb1) > NMS_THR) sup = 1;
            if (!__any(sup)) {
                if (lane == nk)      kb0 = cb;
                if (lane + 32 == nk) kb1 = cb;
                if (lane == 0) {
                    float* ob = outBoxes + ((size_t)img * MAX_FINAL + nk) * 4;
                    ob[0] = cb.x; ob[1] = cb.y; ob[2] = cb.z; ob[3] = cb.w;
                    outMask[(size_t)img * MAX_FINAL + nk] = 1.0f;
                }
                ++nk;
            }
        }

        if (lane == 0) {
            for (int s = nk; s < MAX_FINAL; ++s) {
                float* ob = outBoxes + ((size_t)img * MAX_FINAL + s) * 4;
                ob[0] = 0.f; ob[1] = 0.f; ob[2] = 0.f; ob[3] = 0.f;
                outMask[(size_t)img * MAX_FINAL + s] = 0.0f;
            }
            outCnt[img] = (float)nk;   // count = min(sum(keep), num_final)
        }
    }
}

extern "C" void kernel_launch(void* const* d_in, const int* in_sizes, int n_in,
                              void* d_out, int out_size, void* d_ws, size_t ws_size,
                              hipStream_t stream) {
    const float* boxes  = (const float*)d_in[0];   // [B, N, 4] f32
    const float* scores = (const float*)d_in[1];   // [B, N]    f32
    const int*   nfin   = (const int*)d_in[2];     // [B]       i32
    const int*   shapes = (const int*)d_in[3];     // [B, 2]    i32 (H, W)

    const int B = in_sizes[2];
    const int N = in_sizes[1] / B;                 // 4000 (<= SORT_M)

    size_t shmem = (size_t)N * sizeof(float4) + (size_t)SORT_M * sizeof(unsigned long long);
    nms_per_image_kernel<<<B, BLOCK_T, shmem, stream>>>(
        boxes, scores, nfin, shapes, (float*)d_out, N, B);
}